// SenEmbedding_Loss_48636209660143
// MI455X (gfx1250) — compile-verified
//
#include <hip/hip_runtime.h>
#include <hip/hip_bf16.h>
#include <math.h>

typedef __attribute__((ext_vector_type(2))) float v2f;
typedef __attribute__((ext_vector_type(8))) float v8f;

namespace {
constexpr int S = 64;
constexpr int B = 512;
constexpr int D = 768;
constexpr int P = B * (S - 1);

// d_ws float layout
constexpr int ACC = 0;              // 0:sum_se 1:sum_cosdiff 2:nvalid 3:sum_delta 4:sum_dod 5:K 6:sum_bce
constexpr int NL2 = 16;             // per-row ||l||^2, layout [b*S + s]
constexpr int NT2 = NL2 + S * B;    // per-row ||t||^2
constexpr int DLL = NT2 + S * B;    // raw pair dots l_s . l_{s+1}, layout [b*(S-1) + s]
constexpr int DTT = DLL + P;        // raw pair dots t_s . t_{s+1}
} // namespace

// ---------- helpers ----------

__device__ __forceinline__ float block_sum(float v, float* smem) {
  #pragma unroll
  for (int off = 16; off; off >>= 1) v += __shfl_xor(v, off, 32);
  const int lane = threadIdx.x & 31;
  const int w = threadIdx.x >> 5;
  __syncthreads();
  if (lane == 0) smem[w] = v;
  __syncthreads();
  const int nw = blockDim.x >> 5;
  v = (threadIdx.x < nw) ? smem[threadIdx.x] : 0.0f;
  if (w == 0) {
    #pragma unroll
    for (int off = 4; off; off >>= 1) v += __shfl_xor(v, off, 32);
  }
  return v; // thread 0 holds the block sum
}

__device__ __forceinline__ float pair_cos_from_ws(const float* n2, const float* dotw,
                                                  int b, int s) {
  float na = fmaxf(sqrtf(n2[b * S + s]),     1e-6f);
  float nb = fmaxf(sqrtf(n2[b * S + s + 1]), 1e-6f);
  return dotw[b * (S - 1) + s] / (na * nb);
}

// ---------- kernels ----------

__global__ void k_init(float* __restrict__ ws) {
  if (threadIdx.x < 16) ws[ACC + threadIdx.x] = 0.0f;
}

// One wave per (b, group-of-8 s). Rows 0-7 = logits[s0..s0+7], rows 8-15 = tgt[s0..s0+7].
// Accumulate Gram G = A * A^T with V_WMMA_F32_16X16X4_F32 over K=768.
// 32-bit A (16x4): lanes 0-15 hold row M=lane, K={0,1}; lanes 16-31 hold row M=lane-16, K={2,3}.
// 32-bit B (4x16): same lane placement when B = A^T, so the same v2f is passed as A and B.
// C layout: element (m,n): m<8 -> VGPR m, lane n ; m>=8 -> VGPR m-8, lane 16+n.
__global__ void __launch_bounds__(256)
k_rowstats(const float* __restrict__ logits, const float* __restrict__ tgt,
           const unsigned char* __restrict__ pad, float* __restrict__ ws) {
  float* acc  = ws + ACC;
  float* nl2w = ws + NL2;
  float* nt2w = ws + NT2;
  float* dllw = ws + DLL;
  float* dttw = ws + DTT;

  const int lane = threadIdx.x & 31;
  const int wid  = (blockIdx.x * blockDim.x + threadIdx.x) >> 5; // 0..B*8-1
  const int g = wid & 7;       // s-group
  const int b = wid >> 3;      // batch
  const int r  = lane & 15;    // Gram row for this lane
  const int hi = lane >> 4;    // K sub-slot (0 -> K 0,1 ; 1 -> K 2,3)
  const int s_row = g * 8 + (r & 7);
  const float* base = (r < 8) ? logits : tgt;
  const float* rowp = base + ((size_t)s_row * B + b) * D;

  v8f c = {};
  for (int kb = 0; kb < D; kb += 64) {
    if (kb + 128 <= D) __builtin_prefetch(rowp + kb + 128, 0, 1); // global_prefetch_b8
    #pragma unroll
    for (int kk = 0; kk < 64; kk += 4) {
      v2f a = *(const v2f*)(rowp + kb + kk + 2 * hi);
      c = __builtin_amdgcn_wmma_f32_16x16x4_f32(false, a, false, a, (short)0, c,
                                                false, false);
    }
  }

  // Extract norms / dots from the Gram and accumulate masked MSE/cos terms.
  float red_se = 0.0f, red_cos = 0.0f, red_cnt = 0.0f;
  #pragma unroll
  for (int i = 0; i < 8; ++i) {
    float nl2 = __shfl(c[i], i,      32);  // G[i][i]
    float nt2 = __shfl(c[i], 24 + i, 32);  // G[8+i][8+i]
    float dlt = __shfl(c[i], 8 + i,  32);  // G[i][8+i]
    float dll = __shfl(c[i], i + 1,  32);  // G[i][i+1]      (i<7)
    float dtt = __shfl(c[i], 25 + i, 32);  // G[8+i][8+i+1]  (i<7)
    const int s = g * 8 + i;
    if (lane == 0) {
      nl2w[b * S + s] = nl2;
      nt2w[b * S + s] = nt2;
      if (!pad[b * S + s]) {
        red_cnt += 1.0f;
        red_se  += nl2 + nt2 - 2.0f * dlt;                 // sum((l-t)^2) over D
        float cosv = dlt / (fmaxf(sqrtf(nl2), 1e-8f) * fmaxf(sqrtf(nt2), 1e-8f));
        red_cos += 1.0f - cosv;
      }
      if (i < 7) {
        dllw[b * (S - 1) + s] = dll;
        dttw[b * (S - 1) + s] = dtt;
      }
    }
  }

  // Boundary pair (s0+7, s0+8) crosses the 8-row group: plain VALU dot + wave reduce.
  if (g < 7) {
    const int sA = g * 8 + 7;
    const float* la = logits + ((size_t)sA * B + b) * D;
    const float* lb = la + (size_t)B * D;
    const float* ta = tgt + ((size_t)sA * B + b) * D;
    const float* tb = ta + (size_t)B * D;
    float aL = 0.0f, aT = 0.0f;
    for (int k = lane; k < D; k += 32) {
      aL = fmaf(la[k], lb[k], aL);
      aT = fmaf(ta[k], tb[k], aT);
    }
    #pragma unroll
    for (int off = 16; off; off >>= 1) {
      aL += __shfl_xor(aL, off, 32);
      aT += __shfl_xor(aT, off, 32);
    }
    if (lane == 0) {
      dllw[b * (S - 1) + sA] = aL;
      dttw[b * (S - 1) + sA] = aT;
    }
  }

  if (lane == 0) {
    atomicAdd(&acc[0], red_se);
    atomicAdd(&acc[1], red_cos);
    atomicAdd(&acc[2], red_cnt);
  }
}

__global__ void __launch_bounds__(256)
k_bce(const float* __restrict__ is_pad, const unsigned char* __restrict__ pad,
      float* __restrict__ ws) {
  __shared__ float smem[8];
  const int n = blockIdx.x * blockDim.x + threadIdx.x; // over S*B, layout [s*B + b]
  float v = 0.0f;
  if (n < S * B) {
    const int s = n / B;
    const int b = n % B;
    const float z = is_pad[n];
    const float t = pad[b * S + s] ? 1.0f : 0.0f;
    v = fmaxf(z, 0.0f) - z * t + log1pf(expf(-fabsf(z)));
  }
  const float bs = block_sum(v, smem);
  if (threadIdx.x == 0) atomicAdd(&ws[ACC + 6], bs);
}

// Delta loss + difference-of-differences over consecutive *valid* pairs in b-major
// flat order (equivalent to the reference's stable sort + compaction + masking).
__global__ void __launch_bounds__(256)
k_pairs(const unsigned char* __restrict__ pad, float* __restrict__ ws) {
  __shared__ float smem[8];
  const float* nl2w = ws + NL2;
  const float* nt2w = ws + NT2;
  const float* dllw = ws + DLL;
  const float* dttw = ws + DTT;

  const int j = blockIdx.x * blockDim.x + threadIdx.x;
  float d_delta = 0.0f, d_dod = 0.0f, d_cnt = 0.0f;
  if (j < P) {
    const int b = j / (S - 1);
    const int s = j % (S - 1);
    const bool v = !pad[b * S + s] && !pad[b * S + s + 1];
    if (v) {
      const float dlL = pair_cos_from_ws(nl2w, dllw, b, s);
      const float dlT = pair_cos_from_ws(nt2w, dttw, b, s);
      const float diff = dlL - dlT;
      d_delta = diff * diff;
      d_cnt = 1.0f;
      // find next valid pair in flat order (bounded scan; invalid runs are short)
      int j2 = j + 1;
      for (; j2 < P; ++j2) {
        const int b2 = j2 / (S - 1);
        const int s2 = j2 % (S - 1);
        if (!pad[b2 * S + s2] && !pad[b2 * S + s2 + 1]) break;
      }
      if (j2 < P) {
        const int b2 = j2 / (S - 1);
        const int s2 = j2 % (S - 1);
        const float dlL2 = pair_cos_from_ws(nl2w, dllw, b2, s2);
        const float dlT2 = pair_cos_from_ws(nt2w, dttw, b2, s2);
        const float denL = (dlL != 0.0f) ? dlL : 1e-6f;
        const float denT = (dlT != 0.0f) ? dlT : 1e-6f;
        const float dd = (dlL2 - dlL) / denL - (dlT2 - dlT) / denT;
        d_dod = dd * dd;
      }
    }
  }
  float bs = block_sum(d_delta, smem);
  if (threadIdx.x == 0) atomicAdd(&ws[ACC + 3], bs);
  bs = block_sum(d_dod, smem);
  if (threadIdx.x == 0) atomicAdd(&ws[ACC + 4], bs);
  bs = block_sum(d_cnt, smem);
  if (threadIdx.x == 0) atomicAdd(&ws[ACC + 5], bs);
}

__global__ void k_final(const float* __restrict__ ws,
                        const float* __restrict__ l1, const float* __restrict__ l2,
                        const float* __restrict__ l3, const float* __restrict__ l4,
                        const float* __restrict__ l5, float* __restrict__ out) {
  if (threadIdx.x != 0 || blockIdx.x != 0) return;
  const float sum_se = ws[ACC + 0], sum_cos = ws[ACC + 1], nvalid = ws[ACC + 2];
  const float sum_delta = ws[ACC + 3], sum_dod = ws[ACC + 4], K = ws[ACC + 5];
  const float sum_bce = ws[ACC + 6];
  const float loss_mse   = sum_se / (nvalid * (float)D);
  const float loss_cos   = sum_cos / nvalid;
  const float loss_delta = sum_delta / fmaxf(K, 1.0f);
  const float loss_dod   = sum_dod / fmaxf(K - 1.0f, 1.0f) * 1e-3f;
  const float loss_bce   = sum_bce / (float)(S * B);
  out[0] = l1[0] * loss_mse + l2[0] * loss_cos + l3[0] * loss_delta +
           l4[0] * loss_dod + l5[0] * loss_bce;
}

// ---------- launcher ----------

extern "C" void kernel_launch(void* const* d_in, const int* in_sizes, int n_in,
                              void* d_out, int out_size, void* d_ws, size_t ws_size,
                              hipStream_t stream) {
  (void)in_sizes; (void)n_in; (void)out_size; (void)ws_size;
  const float*         logits = (const float*)d_in[0];
  const float*         is_pad = (const float*)d_in[1];
  const float*         tgt    = (const float*)d_in[2];
  const unsigned char* pad    = (const unsigned char*)d_in[3]; // jnp bool = 1 byte
  const float* l1 = (const float*)d_in[4];
  const float* l2 = (const float*)d_in[5];
  const float* l3 = (const float*)d_in[6];
  const float* l4 = (const float*)d_in[7];
  const float* l5 = (const float*)d_in[8];
  float* ws  = (float*)d_ws;
  float* out = (float*)d_out;

  k_init<<<1, 32, 0, stream>>>(ws);
  // B*8 waves, 8 waves (256 threads) per block
  k_rowstats<<<(B * 8 * 32) / 256, 256, 0, stream>>>(logits, tgt, pad, ws);
  k_bce<<<(S * B + 255) / 256, 256, 0, stream>>>(is_pad, pad, ws);
  k_pairs<<<(P + 255) / 256, 256, 0, stream>>>(pad, ws);
  k_final<<<1, 1, 0, stream>>>(ws, l1, l2, l3, l4, l5, out);
}